// TaskAlignedAssigner_80779744903635
// MI455X (gfx1250) — compile-verified
//
#include <hip/hip_runtime.h>
#include <hip/hip_bf16.h>

#define A_TOT 8400
#define B_TOT 32
#define N_GT  32
#define N_CLS 80
#define K_TOP 13

typedef float v2f __attribute__((ext_vector_type(2)));
typedef float v8f __attribute__((ext_vector_type(8)));

// ---------------------------------------------------------------------------
// K0: zero the per-batch class-count accumulator (must be re-zeroed per call)
// ---------------------------------------------------------------------------
__global__ __launch_bounds__(256) void k_zero(int* __restrict__ colsum) {
  int i = blockIdx.x * 256 + threadIdx.x;
  if (i < B_TOT * N_CLS) colsum[i] = 0;
}

// ---------------------------------------------------------------------------
// K1: bbox_scores[b,n,a] = pd_scores[b,a,labels[b,n]] as one-hot GEMM via
//     V_WMMA_F32_16X16X4_F32 (exact: one-hot A -> each output sums 1 term).
//     One wave = one (batch, 16-anchor tile); D = 32 GTs x 16 anchors via
//     two accumulators. A layout (ISA 16x4 f32): lanes 0-15 hold K=0/1,
//     lanes 16-31 hold K=2/3; M = lane&15. B (4x16): K wraps lane halves.
// ---------------------------------------------------------------------------
__global__ __launch_bounds__(256) void k_gather_wmma(
    const float* __restrict__ pd_scores, const int* __restrict__ gt_labels,
    float* __restrict__ bbox_scores) {
  const int TPB = A_TOT / 16;  // 525 tiles per batch
  int wave = threadIdx.x >> 5;
  int lane = threadIdx.x & 31;
  int tile = blockIdx.x * 8 + wave;       // 2100 blocks * 8 waves = 16800 exact
  int b    = tile / TPB;
  int at   = tile - b * TPB;
  int abase = at * 16;
  int l    = lane & 15;
  int half = lane >> 4;

  int lab0 = gt_labels[b * N_GT + l];        // A-tile rows 0..15  (gts 0..15)
  int lab1 = gt_labels[b * N_GT + 16 + l];   // A-tile rows 0..15  (gts 16..31)

  const float* srow = pd_scores + ((size_t)b * A_TOT + abase + l) * N_CLS;
  __builtin_prefetch(srow, 0, 3);            // -> global_prefetch_b8

  v8f acc0 = {};
  v8f acc1 = {};
  for (int cb = 0; cb < N_CLS; cb += 4) {
    int k0 = cb + half * 2;
    float2 bv = *(const float2*)(srow + k0);
    v2f bm; bm.x = bv.x; bm.y = bv.y;
    v2f a0; a0.x = (lab0 == k0) ? 1.f : 0.f; a0.y = (lab0 == k0 + 1) ? 1.f : 0.f;
    v2f a1; a1.x = (lab1 == k0) ? 1.f : 0.f; a1.y = (lab1 == k0 + 1) ? 1.f : 0.f;
    acc0 = __builtin_amdgcn_wmma_f32_16x16x4_f32(false, a0, false, bm, (short)0, acc0, false, false);
    acc1 = __builtin_amdgcn_wmma_f32_16x16x4_f32(false, a1, false, bm, (short)0, acc1, false, false);
  }
  __builtin_amdgcn_s_wait_tensorcnt(0);  // CDNA5 split-counter wait (no-op here)

  // D layout: lanes 0-15 -> rows 0..7 (vgpr r), lanes 16-31 -> rows 8..15.
  float* o0 = bbox_scores + ((size_t)b * N_GT + half * 8) * A_TOT + abase + l;
  float* o1 = bbox_scores + ((size_t)b * N_GT + 16 + half * 8) * A_TOT + abase + l;
#pragma unroll
  for (int r = 0; r < 8; ++r) {
    o0[(size_t)r * A_TOT] = acc0[r];
    o1[(size_t)r * A_TOT] = acc1[r];
  }
}

// ---------------------------------------------------------------------------
// K2: fused CIoU + in-gt mask + align metric. metrics written in-place over
//     bbox_scores (same index). overlaps + mask_in stored for later passes.
// ---------------------------------------------------------------------------
__global__ __launch_bounds__(256) void k_ciou(
    const float4* __restrict__ pd_bboxes, const float4* __restrict__ gt_bboxes,
    const float2* __restrict__ anc, float* __restrict__ metrics_io,
    float* __restrict__ overlaps, unsigned char* __restrict__ maskin) {
  int a = blockIdx.x * 256 + threadIdx.x;
  int n = blockIdx.y, b = blockIdx.z;
  if (a >= A_TOT) return;
  float4 g = gt_bboxes[b * N_GT + n];
  float4 p = pd_bboxes[(size_t)b * A_TOT + a];
  float2 ap = anc[a];
  const float eps = 1e-7f;
  float w1 = g.z - g.x, h1 = g.w - g.y;
  float w2 = p.z - p.x, h2 = p.w - p.y;
  float iw = fminf(g.z, p.z) - fmaxf(g.x, p.x);
  float ih = fminf(g.w, p.w) - fmaxf(g.y, p.y);
  float inter = fmaxf(iw, 0.f) * fmaxf(ih, 0.f);
  float uni = w1 * h1 + w2 * h2 - inter + eps;
  float iou = inter / uni;
  float cw = fmaxf(g.z, p.z) - fminf(g.x, p.x);
  float ch = fmaxf(g.w, p.w) - fminf(g.y, p.y);
  float c2 = cw * cw + ch * ch + eps;
  float dx = p.x + p.z - g.x - g.z;
  float dy = p.y + p.w - g.y - g.w;
  float rho2 = (dx * dx + dy * dy) * 0.25f;
  float da = atanf(w2 / (h2 + eps)) - atanf(w1 / (h1 + eps));
  float v  = 0.40528473456935108577f * da * da;   // 4/pi^2
  float alpha = v / (v - iou + (1.f + eps));
  float ciou = iou - (rho2 / c2 + v * alpha);
  float ov = fmaxf(ciou, 0.f);
  float dmin = fminf(fminf(ap.x - g.x, ap.y - g.y), fminf(g.z - ap.x, g.w - ap.y));
  unsigned char mi = (dmin > 1e-9f) ? (unsigned char)1 : (unsigned char)0;
  size_t idx = ((size_t)b * N_GT + n) * A_TOT + a;
  float sc = metrics_io[idx];               // bbox_scores from K1
  float o2 = ov * ov;
  overlaps[idx] = ov;
  maskin[idx] = mi;
  metrics_io[idx] = sc * (o2 * o2 * o2) * (float)mi;   // ALPHA=1, BETA=6
}

// ---------------------------------------------------------------------------
// K3: per (b,n) top-13 over A=8400 (row cached in LDS, 13 iterative block
//     argmaxes, tie -> smallest index to match lax.top_k). Writes mask_pos,
//     and ratio[b,n] = pos_overlaps/(pos_align+EPS).
// ---------------------------------------------------------------------------
__global__ __launch_bounds__(256) void k_topk(
    const float* __restrict__ metrics, const float* __restrict__ overlaps,
    const unsigned char* __restrict__ maskin, const float* __restrict__ mask_gt,
    unsigned char* __restrict__ maskpos, float* __restrict__ ratio) {
  __shared__ float vals[A_TOT];
  __shared__ float redv[256];
  __shared__ int   redi[256];
  __shared__ int   sel[K_TOP];
  __shared__ float selv[K_TOP];
  int t = threadIdx.x;
  int bn = blockIdx.x;
  size_t base = (size_t)bn * A_TOT;
  for (int i = t; i < A_TOT; i += 256) { vals[i] = metrics[base + i]; maskpos[base + i] = 0; }
  __syncthreads();
  for (int k = 0; k < K_TOP; ++k) {
    float best = -1.f; int bi = A_TOT;       // metrics >= 0; removed entries = -2
    for (int i = t; i < A_TOT; i += 256) {
      float v = vals[i];
      if (v > best) { best = v; bi = i; }
    }
    redv[t] = best; redi[t] = bi;
    __syncthreads();
    for (int s = 128; s > 0; s >>= 1) {
      if (t < s) {
        float v2 = redv[t + s]; int i2 = redi[t + s];
        if (v2 > redv[t] || (v2 == redv[t] && i2 < redi[t])) { redv[t] = v2; redi[t] = i2; }
      }
      __syncthreads();
    }
    if (t == 0) { sel[k] = redi[0]; selv[k] = redv[0]; vals[redi[0]] = -2.f; }
    __syncthreads();
  }
  if (t == 0) {
    float mg = mask_gt[bn];
    float pa = 0.f, po = 0.f;
    if (mg != 0.f) {
      for (int k = 0; k < K_TOP; ++k) {
        int idx = sel[k];
        unsigned char mi = maskin[base + idx];
        maskpos[base + idx] = mi;            // is_in_topk * mask_in * mask_gt
        pa = fmaxf(pa, selv[k]);
        po = fmaxf(po, overlaps[base + idx] * (float)mi);
      }
    }
    ratio[bn] = po / (pa + 1e-9f);
  }
}

// ---------------------------------------------------------------------------
// K4: per (b,a) column pass: overlap colmax, tgt, fg count, norm, bboxes,
//     per-block class histogram -> global colsum.
// ---------------------------------------------------------------------------
__global__ __launch_bounds__(256) void k_assign(
    const float* __restrict__ overlaps, const float* __restrict__ metrics,
    const unsigned char* __restrict__ maskpos, const int* __restrict__ gt_labels,
    const float4* __restrict__ gt_bboxes, const float* __restrict__ ratio,
    float* __restrict__ out_bboxes, float* __restrict__ out_fg,
    float* __restrict__ out_tgt, float* __restrict__ normv,
    float* __restrict__ fgsum, int* __restrict__ colsum) {
  __shared__ int    slab[N_GT];
  __shared__ float4 sbox[N_GT];
  __shared__ float  srat[N_GT];
  __shared__ int    shist[N_CLS];
  int t = threadIdx.x, b = blockIdx.y;
  if (t < N_GT) {
    slab[t] = gt_labels[b * N_GT + t];
    sbox[t] = gt_bboxes[b * N_GT + t];
    srat[t] = ratio[b * N_GT + t];
  }
  if (t < N_CLS) shist[t] = 0;
  __syncthreads();
  int a = blockIdx.x * 256 + t;
  if (a < A_TOT) {
    size_t base = (size_t)b * N_GT * A_TOT + a;
    float cmax = -3.4e38f;
#pragma unroll 4
    for (int n = 0; n < N_GT; ++n) cmax = fmaxf(cmax, overlaps[base + (size_t)n * A_TOT]);
    int g = 0; float fg = 0.f, nrm = 0.f;
    for (int n = 0; n < N_GT; ++n) {
      size_t idx = base + (size_t)n * A_TOT;
      float ov = overlaps[idx];
      unsigned char mp = maskpos[idx];
      int tg = (mp && ov == cmax) ? n : 0;   // tgt = n * mask_pos * max_mask
      out_tgt[idx] = (float)tg;
      if (tg > g) g = tg;                    // max == value at argmax row
      fg += (float)mp;
      float contrib = mp ? metrics[idx] * srat[n] : 0.f;
      nrm = fmaxf(nrm, contrib);
      atomicAdd(&shist[slab[tg]], 1);        // hard counts per class
    }
    ((float4*)out_bboxes)[(size_t)b * A_TOT + a] = sbox[g];
    out_fg[(size_t)b * A_TOT + a] = (fg != 0.f) ? 1.f : 0.f;  // fg_mask bool
    fgsum[(size_t)b * A_TOT + a]  = fg;
    normv[(size_t)b * A_TOT + a]  = nrm;
  }
  __syncthreads();
  if (t < N_CLS && shist[t]) atomicAdd(&colsum[b * N_CLS + t], shist[t]);
}

// ---------------------------------------------------------------------------
// K5: fg_columns[b] = argmax_c colsum (first max on ties)
// ---------------------------------------------------------------------------
__global__ void k_fgcol(const int* __restrict__ colsum, int* __restrict__ fgcol) {
  int b = blockIdx.x;
  if (threadIdx.x == 0) {
    int best = colsum[b * N_CLS], bi = 0;
    for (int c = 1; c < N_CLS; ++c) {
      int v = colsum[b * N_CLS + c];
      if (v > best) { best = v; bi = c; }
    }
    fgcol[b] = bi;
  }
}

// ---------------------------------------------------------------------------
// K6: target_scores. hard_final[c!=fgcol] = (count_c>0); at fgcol the
//     reference's subtract/trunc/clip collapses to (count_fg + fg_sum != 32).
// ---------------------------------------------------------------------------
__global__ __launch_bounds__(256) void k_scores(
    const float* __restrict__ out_tgt, const int* __restrict__ gt_labels,
    const int* __restrict__ fgcol, const float* __restrict__ normv,
    const float* __restrict__ fgsum, float* __restrict__ out_scores) {
  __shared__ int slab[N_GT];
  int t = threadIdx.x, b = blockIdx.y;
  if (t < N_GT) slab[t] = gt_labels[b * N_GT + t];
  __syncthreads();
  int a = blockIdx.x * 256 + t;
  if (a >= A_TOT) return;
  int fgc = fgcol[b];
  size_t base = (size_t)b * N_GT * A_TOT + a;
  unsigned int m0 = 0, m1 = 0, m2 = 0;
  int cfg = 0;
  for (int n = 0; n < N_GT; ++n) {
    int g = (int)out_tgt[base + (size_t)n * A_TOT];
    int c = slab[g];
    if (c < 32) m0 |= 1u << c;
    else if (c < 64) m1 |= 1u << (c - 32);
    else m2 |= 1u << (c - 64);
    cfg += (c == fgc) ? 1 : 0;
  }
  float nv = normv[(size_t)b * A_TOT + a];
  int fgi = (int)fgsum[(size_t)b * A_TOT + a];
  float* o = out_scores + ((size_t)b * A_TOT + a) * N_CLS;
#pragma unroll
  for (int c = 0; c < N_CLS; ++c) {
    bool h;
    if (c == fgc) h = (cfg + fgi) != N_GT;
    else if (c < 32) h = (m0 >> c) & 1u;
    else if (c < 64) h = (m1 >> (c - 32)) & 1u;
    else h = (m2 >> (c - 64)) & 1u;
    o[c] = h ? nv : 0.f;
  }
}

// ---------------------------------------------------------------------------
extern "C" void kernel_launch(void* const* d_in, const int* in_sizes, int n_in,
                              void* d_out, int out_size, void* d_ws, size_t ws_size,
                              hipStream_t stream) {
  (void)in_sizes; (void)n_in; (void)out_size; (void)ws_size;
  const float* pd_scores = (const float*)d_in[0];   // (B,A,C)
  const float* pd_bboxes = (const float*)d_in[1];   // (B,A,4)
  const float* anc       = (const float*)d_in[2];   // (A,2)
  const int*   gt_labels = (const int*)d_in[3];     // (B,N,1)
  // d_in[4] = gt_probs: unused by the reference computation
  const float* gt_bboxes = (const float*)d_in[5];   // (B,N,4)
  const float* mask_gt   = (const float*)d_in[6];   // (B,N,1)

  // Output layout: bboxes | scores | fg_mask | tgt (all as float)
  float* out_bboxes = (float*)d_out;
  float* out_scores = out_bboxes + (size_t)B_TOT * A_TOT * 4;
  float* out_fg     = out_scores + (size_t)B_TOT * A_TOT * N_CLS;
  float* out_tgt    = out_fg + (size_t)B_TOT * A_TOT;

  // Workspace carve-up (all region sizes are multiples of 256 bytes)
  char* w = (char*)d_ws;
  float* ws_metrics = (float*)w;                 w += (size_t)B_TOT * N_GT * A_TOT * 4;  // bbox_scores -> metrics (in place)
  float* ws_ov      = (float*)w;                 w += (size_t)B_TOT * N_GT * A_TOT * 4;
  unsigned char* ws_mi = (unsigned char*)w;      w += (size_t)B_TOT * N_GT * A_TOT;
  unsigned char* ws_mp = (unsigned char*)w;      w += (size_t)B_TOT * N_GT * A_TOT;
  float* ws_ratio   = (float*)w;                 w += (size_t)B_TOT * N_GT * 4;
  float* ws_norm    = (float*)w;                 w += (size_t)B_TOT * A_TOT * 4;
  float* ws_fgsum   = (float*)w;                 w += (size_t)B_TOT * A_TOT * 4;
  int*   ws_colsum  = (int*)w;                   w += (size_t)B_TOT * N_CLS * 4;
  int*   ws_fgcol   = (int*)w;

  const int ABLK = (A_TOT + 255) / 256;  // 33

  k_zero<<<(B_TOT * N_CLS + 255) / 256, 256, 0, stream>>>(ws_colsum);
  k_gather_wmma<<<(B_TOT * (A_TOT / 16)) / 8, 256, 0, stream>>>(pd_scores, gt_labels, ws_metrics);
  k_ciou<<<dim3(ABLK, N_GT, B_TOT), 256, 0, stream>>>(
      (const float4*)pd_bboxes, (const float4*)gt_bboxes, (const float2*)anc,
      ws_metrics, ws_ov, ws_mi);
  k_topk<<<B_TOT * N_GT, 256, 0, stream>>>(ws_metrics, ws_ov, ws_mi, mask_gt, ws_mp, ws_ratio);
  k_assign<<<dim3(ABLK, B_TOT), 256, 0, stream>>>(
      ws_ov, ws_metrics, ws_mp, gt_labels, (const float4*)gt_bboxes, ws_ratio,
      out_bboxes, out_fg, out_tgt, ws_norm, ws_fgsum, ws_colsum);
  k_fgcol<<<B_TOT, 32, 0, stream>>>(ws_colsum, ws_fgcol);
  k_scores<<<dim3(ABLK, B_TOT), 256, 0, stream>>>(
      out_tgt, gt_labels, ws_fgcol, ws_norm, ws_fgsum, out_scores);
}